// ImageLaplacianBuilder_67250597921256
// MI455X (gfx1250) — compile-verified
//
#include <hip/hip_runtime.h>
#include <stdint.h>

#define H 512
#define W 512
#define K 112
#define TB 16                 // pixels per tile side
#define HALO 6
#define TILE (TB + 2*HALO)    // 28
#define ROWF (TILE*3)         // 84 floats per LDS tile row

typedef __attribute__((ext_vector_type(4))) float        v4f;
typedef __attribute__((ext_vector_type(4))) int          v4i;
typedef __attribute__((ext_vector_type(4))) unsigned int v4u;
typedef __attribute__((ext_vector_type(8))) int          v8i;

// ---------------- compile-time offset table (matches reference sorted order) ----
struct Tab { int dydx[K]; float w[K]; };

constexpr Tab build_offsets() {
  Tab t{};
  const float rw[6] = {1.0f, 0.6f, 0.4f, 0.3f, 0.2f, 0.1f};
  int idx = 0;
  for (int dy = -6; dy <= 6; ++dy)
    for (int dx = -6; dx <= 6; ++dx) {
      int d2 = dy*dy + dx*dx;
      if (d2 == 0 || d2 > 36) continue;
      float w = 0.0f;
      for (int r = 1; r <= 6; ++r)
        if (d2 <= r*r) w += rw[r-1];
      t.dydx[idx] = (int)((((unsigned)dy & 0xFFFFu) << 16) | ((unsigned)dx & 0xFFFFu));
      t.w[idx]    = w;
      ++idx;
    }
  return t;
}

__constant__ Tab c_tab = build_offsets();

// ---------------- pass 1: image max (for [0,255] vs [0,1] scale detection) -----
__global__ __launch_bounds__(256) void max_reduce(const float* __restrict__ x, int n,
                                                  int* __restrict__ out) {
  float m = 0.0f;
  for (int i = blockIdx.x * blockDim.x + threadIdx.x; i < n;
       i += gridDim.x * blockDim.x)
    m = fmaxf(m, x[i]);
#pragma unroll
  for (int o = 16; o > 0; o >>= 1)
    m = fmaxf(m, __shfl_xor(m, o, 32));
  if ((threadIdx.x & 31) == 0)
    atomicMax(out, __float_as_int(m));   // int-bit compare == float compare (x >= 0)
}

// ---------------- pass 2: Laplacian edges -------------------------------------
__global__ __launch_bounds__(256) void lap_main(const float* __restrict__ img,
                                                const int* __restrict__ wsmax,
                                                float* __restrict__ vals,
                                                int* __restrict__ cols,
                                                float* __restrict__ degs) {
  __shared__ float tile[TILE * ROWF];   // 28x28 px halo tile, 3ch f32 = 9408 B

  const int bx = blockIdx.x, by = blockIdx.y;
  // clamp DMA origin so the full 28x28 window stays inside the image
  const int xs = min(max(bx * TB - HALO, 0), W - TILE);
  const int ys = min(max(by * TB - HALO, 0), H - TILE);
  const int tid = threadIdx.y * TB + threadIdx.x;

  if (tid < 32) {  // wave 0 drives the Tensor Data Mover for the whole block
    unsigned long long ga =
        (unsigned long long)(uintptr_t)(img + ((size_t)ys * W + xs) * 3);
    unsigned lds_off = (unsigned)(uintptr_t)(void*)tile;  // flat LDS addr[31:0]

    // D# group 0: count=1 | lds_addr | global_addr[56:0] | type=2
    v4u g0 = { 1u,
               lds_off,
               (unsigned)(ga & 0xFFFFFFFFull),
               (unsigned)((ga >> 32) & 0x1FFFFFFull) | 0x80000000u };
    // D# group 1: data_size=4B; tensor 1536x512 f32; tile 84x28; stride 1536
    v8i g1 = { (int)(2u << 16),                                   // data_size=2 (4B)
               (int)(((unsigned)(W * 3) & 0xFFFFu) << 16),        // tensor_dim0 lo16
               (int)(((unsigned)(W * 3) >> 16) | ((unsigned)H << 16)),
               (int)(((unsigned)H >> 16) | ((unsigned)ROWF << 16)), // tile_dim0=84
               (int)TILE,                                         // tile_dim1=28
               (int)(W * 3),                                      // dim0_stride=1536
               0, 0 };
    v4i gz4 = {0, 0, 0, 0};
    v8i gz8 = {0, 0, 0, 0, 0, 0, 0, 0};
    __builtin_amdgcn_tensor_load_to_lds(g0, g1, gz4, gz4, gz8, 0);
    __builtin_amdgcn_s_wait_tensorcnt(0);
  }
  __syncthreads();

  const int gx  = bx * TB + threadIdx.x;
  const int gy  = by * TB + threadIdx.y;
  const int pix = gy * W + gx;

  const float* own = &tile[(gy - ys) * ROWF + (gx - xs) * 3];
  const float c0 = own[0], c1 = own[1], c2 = own[2];

  const float mx    = __int_as_float(wsmax[0]);
  const float scale = (mx > 1.0f) ? (1.0f / 255.0f) : 1.0f;
  // exp(-scale*||d||/0.15) = exp2(||d|| * (-scale*log2(e)/0.15))
  const float kexp  = scale * -9.6179669392597560f;

  v4f* __restrict__ pv = (v4f*)(vals + (size_t)pix * K);
  v4i* __restrict__ pc = (v4i*)(cols + (size_t)pix * K);

  float deg = 0.0f;
#pragma unroll 2
  for (int k = 0; k < K; k += 4) {
    v4f vv; v4i cc;
#pragma unroll
    for (int j = 0; j < 4; ++j) {
      const int   e  = c_tab.dydx[k + j];
      const float w  = c_tab.w[k + j];
      const int   ny = gy + (e >> 16);
      const int   nx = gx + (int)(short)e;
      const bool valid = ((unsigned)ny < (unsigned)H) & ((unsigned)nx < (unsigned)W);
      // clamped into the loaded window; exact for all valid neighbors
      const int ly = min(max(ny - ys, 0), TILE - 1);
      const int lx = min(max(nx - xs, 0), TILE - 1);
      const float* nb = &tile[ly * ROWF + lx * 3];
      const float d0 = c0 - nb[0];
      const float d1 = c1 - nb[1];
      const float d2 = c2 - nb[2];
      const float cd = __builtin_amdgcn_sqrtf(
          __builtin_fmaf(d0, d0, __builtin_fmaf(d1, d1, d2 * d2)));
      const float ew = w * __builtin_amdgcn_exp2f(cd * kexp);
      ((float*)&vv)[j] = valid ? -ew : 0.0f;
      ((int*)&cc)[j]   = valid ? (ny * W + nx) : pix;
      deg += valid ? ew : 0.0f;
    }
    __builtin_nontemporal_store(vv, pv + (k >> 2));
    __builtin_nontemporal_store(cc, pc + (k >> 2));
  }
  __builtin_nontemporal_store(deg, degs + pix);
}

// ---------------- launch ------------------------------------------------------
extern "C" void kernel_launch(void* const* d_in, const int* in_sizes, int n_in,
                              void* d_out, int out_size, void* d_ws, size_t ws_size,
                              hipStream_t stream) {
  (void)in_sizes; (void)n_in; (void)out_size; (void)ws_size;
  const float* img = (const float*)d_in[0];
  const size_t NK = (size_t)H * W * K;
  float* vals = (float*)d_out;
  int*   cols = (int*)d_out + NK;
  float* degs = (float*)d_out + 2 * NK;

  (void)hipMemsetAsync(d_ws, 0, sizeof(int), stream);
  max_reduce<<<192, 256, 0, stream>>>(img, H * W * 3, (int*)d_ws);

  dim3 blk(TB, TB), grd(W / TB, H / TB);
  lap_main<<<grd, blk, 0, stream>>>(img, (const int*)d_ws, vals, cols, degs);
}